// BSCAUnrolled_69346541961979
// MI455X (gfx1250) — compile-verified
//
#include <hip/hip_runtime.h>
#include <math.h>

#define B_DIM 16
#define E_DIM 256
#define T_DIM 4096
#define F_DIM 512
#define RANK 10
#define NUM_ITER 5

typedef __attribute__((ext_vector_type(2))) float v2f;
typedef __attribute__((ext_vector_type(8))) float v8f;

// D = A(16x4) * B(4x16) + C, fp32 WMMA (CDNA5)
__device__ __forceinline__ v8f wmma4f32(v2f a, v2f b, v8f c) {
  return __builtin_amdgcn_wmma_f32_16x16x4_f32(false, a, false, b, (short)0, c,
                                               false, false);
}

// ---- gfx1250 async global->LDS copies (ASYNCcnt-tracked DMA) ---------------
__device__ __forceinline__ void async_ld_b128(unsigned lds_off,
                                              const void* gaddr) {
  asm volatile("global_load_async_to_lds_b128 %0, %1, off" ::"v"(lds_off),
               "v"(gaddr)
               : "memory");
}
__device__ __forceinline__ void async_ld_b32(unsigned lds_off,
                                             const void* gaddr) {
  asm volatile("global_load_async_to_lds_b32 %0, %1, off" ::"v"(lds_off),
               "v"(gaddr)
               : "memory");
}
template <int N>
__device__ __forceinline__ void wait_asynccnt() {
  asm volatile("s_wait_asynccnt %0" ::"i"(N) : "memory");
}
__device__ __forceinline__ unsigned lds_addr(const void* p) {
  // low 32 bits of a generic pointer to __shared__ are the LDS byte offset
  return (unsigned)(unsigned long long)p;
}

// ---------------------------------------------------------------------------
// elementwise init: X = Omega * Y
// ---------------------------------------------------------------------------
__global__ void init_X_kernel(const float* __restrict__ Y,
                              const float* __restrict__ Om,
                              float* __restrict__ X, long n) {
  long i = (long)blockIdx.x * blockDim.x + threadIdx.x;
  if (i < n) X[i] = Om[i] * Y[i];
}

// ---------------------------------------------------------------------------
// lhs = M^T M + regp*I (M: n x RANK row-major, per batch), then 10x10 inverse
// ---------------------------------------------------------------------------
__global__ void lhs_inv_kernel(const float* __restrict__ M, int n,
                               const float* __restrict__ lam_log,
                               const float* __restrict__ nu_log, int k,
                               float* __restrict__ invOut) {
  __shared__ float lhs[RANK * RANK];
  __shared__ float aug[RANK][2 * RANK];
  int b = blockIdx.x;
  int tid = threadIdx.x;
  const float* Mb = M + (long)b * n * RANK;
  if (tid < RANK * RANK) {
    int i = tid / RANK, j = tid % RANK;
    float s = 0.f;
    for (int r = 0; r < n; ++r) s += Mb[r * RANK + i] * Mb[r * RANK + j];
    float regp = expf(lam_log[k] - nu_log[k]);
    if (i == j) s += regp;
    lhs[tid] = s;
  }
  __syncthreads();
  if (tid == 0) {
    for (int i = 0; i < RANK; ++i)
      for (int j = 0; j < RANK; ++j) {
        aug[i][j] = lhs[i * RANK + j];
        aug[i][RANK + j] = (i == j) ? 1.f : 0.f;
      }
    for (int c = 0; c < RANK; ++c) {
      float inv = 1.f / aug[c][c];
      for (int j = 0; j < 2 * RANK; ++j) aug[c][j] *= inv;
      for (int r2 = 0; r2 < RANK; ++r2) {
        if (r2 == c) continue;
        float f = aug[r2][c];
        for (int j = 0; j < 2 * RANK; ++j) aug[r2][j] -= f * aug[c][j];
      }
    }
  }
  __syncthreads();
  if (tid < RANK * RANK)
    invOut[(long)b * RANK * RANK + tid] = aug[tid / RANK][RANK + tid % RANK];
}

// ---------------------------------------------------------------------------
// P[b,e,:] = (X[b,e,:] @ Q[b]) @ invQ[b]   (one block per (b,e) row)
// ---------------------------------------------------------------------------
__global__ void P_update_kernel(const float* __restrict__ X,
                                const float* __restrict__ Q,
                                const float* __restrict__ invQ,
                                float* __restrict__ P) {
  int b = blockIdx.x / E_DIM;
  int e = blockIdx.x % E_DIM;
  int tid = threadIdx.x;
  const float* xr = X + ((long)b * E_DIM + e) * T_DIM;
  const float* qb = Q + (long)b * T_DIM * RANK;
  float acc[RANK];
#pragma unroll
  for (int j = 0; j < RANK; ++j) acc[j] = 0.f;
  for (int t = tid; t < T_DIM; t += 256) {
    float xv = xr[t];
#pragma unroll
    for (int j = 0; j < RANK; ++j) acc[j] += xv * qb[t * RANK + j];
  }
  __shared__ float sm[256][RANK];
#pragma unroll
  for (int j = 0; j < RANK; ++j) sm[tid][j] = acc[j];
  __syncthreads();
  for (int s = 128; s > 0; s >>= 1) {
    if (tid < s) {
#pragma unroll
      for (int j = 0; j < RANK; ++j) sm[tid][j] += sm[tid + s][j];
    }
    __syncthreads();
  }
  if (tid < RANK) {
    const float* iq = invQ + (long)b * RANK * RANK;
    float s = 0.f;
#pragma unroll
    for (int j = 0; j < RANK; ++j) s += sm[0][j] * iq[j * RANK + tid];
    P[((long)b * E_DIM + e) * RANK + tid] = s;
  }
}

// ---------------------------------------------------------------------------
// Q[b,t,:] = (X[b,:,t]^T @ P[b]) @ invP[b]   (grid: (T/256, B), 256 thr)
// ---------------------------------------------------------------------------
__global__ void Q_update_kernel(const float* __restrict__ X,
                                const float* __restrict__ P,
                                const float* __restrict__ invP,
                                float* __restrict__ Q) {
  int b = blockIdx.y;
  int tid = threadIdx.x;
  int t = blockIdx.x * 256 + tid;
  __shared__ float Ps[E_DIM * RANK];
  __shared__ float ip[RANK * RANK];
  for (int i = tid; i < E_DIM * RANK; i += 256)
    Ps[i] = P[(long)b * E_DIM * RANK + i];
  if (tid < RANK * RANK) ip[tid] = invP[(long)b * RANK * RANK + tid];
  __syncthreads();
  float acc[RANK];
#pragma unroll
  for (int j = 0; j < RANK; ++j) acc[j] = 0.f;
  const float* xb = X + (long)b * E_DIM * T_DIM + t;
  for (int e = 0; e < E_DIM; ++e) {
    float xv = xb[(long)e * T_DIM];
#pragma unroll
    for (int j = 0; j < RANK; ++j) acc[j] += xv * Ps[e * RANK + j];
  }
  float* qo = Q + ((long)b * T_DIM + t) * RANK;
#pragma unroll
  for (int i = 0; i < RANK; ++i) {
    float s = 0.f;
#pragma unroll
    for (int j = 0; j < RANK; ++j) s += acc[j] * ip[j * RANK + i];
    qo[i] = s;
  }
}

// ---------------------------------------------------------------------------
// GEMM tiling: 64x64 tile, BK=32, 256 threads (8 waves), 2 subtiles/wave,
// double-buffered async global->LDS staging.
// ---------------------------------------------------------------------------
#define AS_LD 36
#define BS_LD 68

// A_scale = (R*R)^T @ (Omega*Omega):  M=F(512), N=T(4096), K=E(256)
__global__ void ascale_kernel(const float* __restrict__ R,
                              const float* __restrict__ Om,
                              float* __restrict__ Ascale) {
  __shared__ float As[2][64][AS_LD];
  __shared__ float Bs[2][32][BS_LD];
  int b = blockIdx.z, f0 = blockIdx.y * 64, t0 = blockIdx.x * 64;
  int tid = threadIdx.x;
  int wave = tid >> 5, lane = tid & 31, hf = lane >> 4, lrow = lane & 15;
  int msub = (wave >> 2) * 16, nsub = (wave & 3) * 16;
  v8f acc0 = {0.f, 0.f, 0.f, 0.f, 0.f, 0.f, 0.f, 0.f};
  v8f acc1 = acc0;
  const float* Rb = R + (long)b * E_DIM * F_DIM;
  const float* Ob = Om + (long)b * E_DIM * T_DIM;
  // load geometry: As[m][k] = R[(k0+k)*F + f0+m] (transposed, b32 per elem)
  int am = tid & 63, akb = (tid >> 6) * 8;
  int bk = tid >> 3, bnq = (tid & 7) * 8;
  const float* aSrc = Rb + f0 + am;             // + (k0+akb+i)*F
  const float* bSrc = Ob + t0 + bnq;            // + (k0+bk)*T
  unsigned asO[2], bsO[2];
#pragma unroll
  for (int s = 0; s < 2; ++s) {
    asO[s] = lds_addr(&As[s][am][akb]);
    bsO[s] = lds_addr(&Bs[s][bk][bnq]);
  }
  const int NC = E_DIM / 32;  // 8 chunks
  auto issue = [&](int c, int s) {
    const float* a = aSrc + (long)(c * 32 + akb) * F_DIM;
#pragma unroll
    for (int i = 0; i < 8; ++i)
      async_ld_b32(asO[s] + 4u * i, a + (long)i * F_DIM);
    const float* bsc = bSrc + (long)(c * 32 + bk) * T_DIM;
    async_ld_b128(bsO[s], bsc);
    async_ld_b128(bsO[s] + 16u, bsc + 4);
  };
  issue(0, 0);
  for (int c = 0; c < NC; ++c) {
    if (c + 1 < NC) {
      issue(c + 1, (c + 1) & 1);
      wait_asynccnt<10>();
    } else {
      wait_asynccnt<0>();
    }
    __syncthreads();
    const float(*Ab)[AS_LD] = As[c & 1];
    const float(*Bb)[BS_LD] = Bs[c & 1];
#pragma unroll
    for (int kk = 0; kk < 32; kk += 4) {
      v2f a0 = *(const v2f*)&Ab[msub + lrow][kk + 2 * hf];
      v2f a1 = *(const v2f*)&Ab[msub + 32 + lrow][kk + 2 * hf];
      v2f bb;
      bb.x = Bb[kk + 2 * hf][nsub + lrow];
      bb.y = Bb[kk + 2 * hf + 1][nsub + lrow];
      a0 = a0 * a0;  // (R*R)^T
      a1 = a1 * a1;
      bb = bb * bb;  // Omega*Omega
      acc0 = wmma4f32(a0, bb, acc0);
      acc1 = wmma4f32(a1, bb, acc1);
    }
    __syncthreads();
  }
  float* outb = Ascale + (long)b * F_DIM * T_DIM;
  int col = t0 + nsub + lrow;
#pragma unroll
  for (int v = 0; v < 8; ++v) {
    int r0 = f0 + msub + v + 8 * hf;
    outb[(long)r0 * T_DIM + col] = acc0[v];
    outb[(long)(r0 + 32) * T_DIM + col] = acc1[v];
  }
}

// RA = R @ A, then X/resid update fused.  M=E(256), N=T(4096), K=F(512)
__global__ void fusedX_kernel(const float* __restrict__ R,
                              const float* __restrict__ Aan,
                              const float* __restrict__ Y,
                              const float* __restrict__ Om,
                              const float* __restrict__ P,
                              const float* __restrict__ Q,
                              const float* __restrict__ nu_log, int k,
                              float* __restrict__ X,
                              float* __restrict__ resid) {
  __shared__ float As[2][64][AS_LD];
  __shared__ float Bs[2][32][BS_LD];
  __shared__ float Pt[64 * RANK];
  __shared__ float Qt[64 * RANK];
  int b = blockIdx.z, e0 = blockIdx.y * 64, t0 = blockIdx.x * 64;
  int tid = threadIdx.x;
  int wave = tid >> 5, lane = tid & 31, hf = lane >> 4, lrow = lane & 15;
  int msub = (wave >> 2) * 16, nsub = (wave & 3) * 16;
  v8f acc0 = {0.f, 0.f, 0.f, 0.f, 0.f, 0.f, 0.f, 0.f};
  v8f acc1 = acc0;
  const float* Rb = R + (long)b * E_DIM * F_DIM;
  const float* Ab = Aan + (long)b * F_DIM * T_DIM;
  int am = tid >> 2, akq = (tid & 3) * 8;  // As row-major: R[(e0+m)*F + k]
  int bk = tid >> 3, bnq = (tid & 7) * 8;  // Bs: A[(k0+k)*T + t0+n]
  const float* aSrc = Rb + (long)(e0 + am) * F_DIM + akq;  // + k0
  const float* bSrc = Ab + (long)bk * T_DIM + t0 + bnq;    // + k0*T
  unsigned asO[2], bsO[2];
#pragma unroll
  for (int s = 0; s < 2; ++s) {
    asO[s] = lds_addr(&As[s][am][akq]);
    bsO[s] = lds_addr(&Bs[s][bk][bnq]);
  }
  const int NC = F_DIM / 32;  // 16 chunks
  auto issue = [&](int c, int s) {
    const float* a = aSrc + c * 32;
    async_ld_b128(asO[s], a);
    async_ld_b128(asO[s] + 16u, a + 4);
    const float* bsc = bSrc + (long)c * 32 * T_DIM;
    async_ld_b128(bsO[s], bsc);
    async_ld_b128(bsO[s] + 16u, bsc + 4);
  };
  issue(0, 0);
  for (int c = 0; c < NC; ++c) {
    if (c + 1 < NC) {
      issue(c + 1, (c + 1) & 1);
      wait_asynccnt<4>();
    } else {
      wait_asynccnt<0>();
    }
    __syncthreads();
    const float(*Asb)[AS_LD] = As[c & 1];
    const float(*Bsb)[BS_LD] = Bs[c & 1];
#pragma unroll
    for (int kk = 0; kk < 32; kk += 4) {
      v2f a0 = *(const v2f*)&Asb[msub + lrow][kk + 2 * hf];
      v2f a1 = *(const v2f*)&Asb[msub + 32 + lrow][kk + 2 * hf];
      v2f bb;
      bb.x = Bsb[kk + 2 * hf][nsub + lrow];
      bb.y = Bsb[kk + 2 * hf + 1][nsub + lrow];
      acc0 = wmma4f32(a0, bb, acc0);
      acc1 = wmma4f32(a1, bb, acc1);
    }
    __syncthreads();
  }
  // stage P rows e0..e0+63 and Q rows t0..t0+63 for the rank-10 epilogue
  for (int i = tid; i < 64 * RANK; i += 256)
    Pt[i] = P[((long)b * E_DIM + e0) * RANK + i];
  for (int i = tid; i < 64 * RANK; i += 256)
    Qt[i] = Q[((long)b * T_DIM + t0) * RANK + i];
  __syncthreads();
  float nu = expf(nu_log[k]);
  int coll = nsub + lrow;
  long col = t0 + coll;
#pragma unroll
  for (int s = 0; s < 2; ++s) {
    v8f acc = s ? acc1 : acc0;
    int mb = msub + s * 32;
#pragma unroll
    for (int v = 0; v < 8; ++v) {
      int ml = mb + v + 8 * hf;
      long idx = ((long)b * E_DIM + e0 + ml) * T_DIM + col;
      float om = Om[idx], y = Y[idx], ra = acc[v];
      float pq = 0.f;
#pragma unroll
      for (int j = 0; j < RANK; ++j)
        pq += Pt[ml * RANK + j] * Qt[coll * RANK + j];
      float err = om * (y - ra);
      float xv = (om * err + nu * pq) / (om * om + nu);
      X[idx] = xv;
      resid[idx] = om * (y - xv - ra);
    }
  }
}

// G = R^T @ resid, then soft-threshold + convex step on A (in-place on d_out)
// M=F(512), N=T(4096), K=E(256)
__global__ void A_update_kernel(const float* __restrict__ R,
                                const float* __restrict__ resid,
                                const float* __restrict__ Ascale,
                                const float* __restrict__ mu_log,
                                const float* __restrict__ gamma_logit, int k,
                                float* __restrict__ Aan) {
  __shared__ float As[2][64][AS_LD];
  __shared__ float Bs[2][32][BS_LD];
  int b = blockIdx.z, f0 = blockIdx.y * 64, t0 = blockIdx.x * 64;
  int tid = threadIdx.x;
  int wave = tid >> 5, lane = tid & 31, hf = lane >> 4, lrow = lane & 15;
  int msub = (wave >> 2) * 16, nsub = (wave & 3) * 16;
  v8f acc0 = {0.f, 0.f, 0.f, 0.f, 0.f, 0.f, 0.f, 0.f};
  v8f acc1 = acc0;
  const float* Rb = R + (long)b * E_DIM * F_DIM;
  const float* resb = resid + (long)b * E_DIM * T_DIM;
  int am = tid & 63, akb = (tid >> 6) * 8;  // As transposed (b32 per elem)
  int bk = tid >> 3, bnq = (tid & 7) * 8;
  const float* aSrc = Rb + f0 + am;
  const float* bSrc = resb + t0 + bnq;
  unsigned asO[2], bsO[2];
#pragma unroll
  for (int s = 0; s < 2; ++s) {
    asO[s] = lds_addr(&As[s][am][akb]);
    bsO[s] = lds_addr(&Bs[s][bk][bnq]);
  }
  const int NC = E_DIM / 32;  // 8 chunks
  auto issue = [&](int c, int s) {
    const float* a = aSrc + (long)(c * 32 + akb) * F_DIM;
#pragma unroll
    for (int i = 0; i < 8; ++i)
      async_ld_b32(asO[s] + 4u * i, a + (long)i * F_DIM);
    const float* bsc = bSrc + (long)(c * 32 + bk) * T_DIM;
    async_ld_b128(bsO[s], bsc);
    async_ld_b128(bsO[s] + 16u, bsc + 4);
  };
  issue(0, 0);
  for (int c = 0; c < NC; ++c) {
    if (c + 1 < NC) {
      issue(c + 1, (c + 1) & 1);
      wait_asynccnt<10>();
    } else {
      wait_asynccnt<0>();
    }
    __syncthreads();
    const float(*Asb)[AS_LD] = As[c & 1];
    const float(*Bsb)[BS_LD] = Bs[c & 1];
#pragma unroll
    for (int kk = 0; kk < 32; kk += 4) {
      v2f a0 = *(const v2f*)&Asb[msub + lrow][kk + 2 * hf];
      v2f a1 = *(const v2f*)&Asb[msub + 32 + lrow][kk + 2 * hf];
      v2f bb;
      bb.x = Bsb[kk + 2 * hf][nsub + lrow];
      bb.y = Bsb[kk + 2 * hf + 1][nsub + lrow];
      acc0 = wmma4f32(a0, bb, acc0);
      acc1 = wmma4f32(a1, bb, acc1);
    }
    __syncthreads();
  }
  float mu = expf(mu_log[k]);
  float g = 1.f / (1.f + expf(-gamma_logit[k]));
  long col = t0 + nsub + lrow;
#pragma unroll
  for (int s = 0; s < 2; ++s) {
    v8f acc = s ? acc1 : acc0;
    int mb = msub + s * 32;
#pragma unroll
    for (int v = 0; v < 8; ++v) {
      int f = f0 + mb + v + 8 * hf;
      long idx = ((long)b * F_DIM + f) * T_DIM + col;
      float as = Ascale[idx];
      bool noobs = (as == 0.f);
      float safe = as + (noobs ? 1.f : 0.f);
      float aold = Aan[idx];
      float ba = aold + acc[v] / safe;
      float ad = fmaxf(fabsf(ba) - mu / safe, 0.f);
      ad = (ba < 0.f) ? -ad : ad;
      if (noobs) ad = 0.f;
      Aan[idx] = aold + g * (ad - aold);
    }
  }
}

// ---------------------------------------------------------------------------
extern "C" void kernel_launch(void* const* d_in, const int* in_sizes, int n_in,
                              void* d_out, int out_size, void* d_ws,
                              size_t ws_size, hipStream_t stream) {
  const float* Y = (const float*)d_in[0];
  const float* R = (const float*)d_in[1];
  const float* Om = (const float*)d_in[2];
  // d_in[3] = P0 is never read by the recurrence (P is overwritten first)
  const float* Q0 = (const float*)d_in[4];
  const float* lam_log = (const float*)d_in[5];
  const float* mu_log = (const float*)d_in[6];
  const float* nu_log = (const float*)d_in[7];
  const float* gamma_logit = (const float*)d_in[8];
  float* A = (float*)d_out;  // (B,F,T)

  const long BET = (long)B_DIM * E_DIM * T_DIM;
  const long BFT = (long)B_DIM * F_DIM * T_DIM;
  float* ws = (float*)d_ws;
  float* X = ws;                      // BET
  float* resid = X + BET;             // BET
  float* Ascale = resid + BET;        // BFT
  float* Q = Ascale + BFT;            // B*T*RANK
  float* P = Q + (long)B_DIM * T_DIM * RANK;       // B*E*RANK
  float* invQ = P + (long)B_DIM * E_DIM * RANK;    // B*RANK*RANK
  float* invP = invQ + (long)B_DIM * RANK * RANK;  // B*RANK*RANK

  hipMemsetAsync(A, 0, BFT * sizeof(float), stream);
  hipMemcpyAsync(Q, Q0, (size_t)B_DIM * T_DIM * RANK * sizeof(float),
                 hipMemcpyDeviceToDevice, stream);
  init_X_kernel<<<(int)((BET + 255) / 256), 256, 0, stream>>>(Y, Om, X, BET);
  ascale_kernel<<<dim3(T_DIM / 64, F_DIM / 64, B_DIM), 256, 0, stream>>>(
      R, Om, Ascale);

  for (int k = 0; k < NUM_ITER; ++k) {
    lhs_inv_kernel<<<B_DIM, 128, 0, stream>>>(Q, T_DIM, lam_log, nu_log, k,
                                              invQ);
    P_update_kernel<<<B_DIM * E_DIM, 256, 0, stream>>>(X, Q, invQ, P);
    lhs_inv_kernel<<<B_DIM, 128, 0, stream>>>(P, E_DIM, lam_log, nu_log, k,
                                              invP);
    Q_update_kernel<<<dim3(T_DIM / 256, B_DIM), 256, 0, stream>>>(X, P, invP,
                                                                  Q);
    fusedX_kernel<<<dim3(T_DIM / 64, E_DIM / 64, B_DIM), 256, 0, stream>>>(
        R, A, Y, Om, P, Q, nu_log, k, X, resid);
    A_update_kernel<<<dim3(T_DIM / 64, F_DIM / 64, B_DIM), 256, 0, stream>>>(
        R, resid, Ascale, mu_log, gamma_logit, k, A);
  }
}